// TimeRestrictedSelfAttention_18915035972190
// MI455X (gfx1250) — compile-verified
//
#include <hip/hip_runtime.h>
#include <hip/hip_bf16.h>
#include <math.h>

// ---------------------------------------------------------------------------
// Problem constants (from reference)
// ---------------------------------------------------------------------------
#define IN_DIM   512
#define OUT_DIM  512
#define KEY_DIM  512
#define FILT     32
#define STRIDE_T 4
#define BATCH    8
#define SEQ_T    2048
#define T_OUT    505            // (2048 - 32)/4 + 1
#define MTOT     (BATCH * SEQ_T) // 16384 rows for the qkv GEMMs

typedef __attribute__((ext_vector_type(2))) float v2f;
typedef __attribute__((ext_vector_type(8))) float v8f;

// ---------------------------------------------------------------------------
// GEMM:  Out[M x 512] = X[M x 512] @ W[512 x 512] + bias
// Block: 256 threads = 8 wave32s. Block tile: 16 (M) x 128 (N).
// Each wave computes one 16x16 tile with V_WMMA_F32_16X16X4_F32, K stepped by 4.
// X strip (16x512) staged in LDS (padded stride to avoid bank conflicts).
// ---------------------------------------------------------------------------
#define XS_STRIDE 516  // 516 % 64 == 4 -> 16 rows land on distinct banks

__global__ __launch_bounds__(256) void qkv_gemm_kernel(
    const float* __restrict__ X,
    const float* __restrict__ W,
    const float* __restrict__ bias,
    float* __restrict__ Out)
{
    __shared__ __align__(16) float xs[16 * XS_STRIDE];

    const int tid      = threadIdx.x;
    const int wave     = tid >> 5;          // 0..7
    const int lane     = tid & 31;          // wave32
    const int row_base = blockIdx.x * 16;   // M tile
    const int col_base = blockIdx.y * 128 + wave * 16; // N tile per wave

    // --- cooperative, coalesced load of the 16x512 X strip into LDS ---------
    // 8192 floats = 2048 float4; 256 threads -> 8 float4 each. No divergence.
    {
        const float4* xb = (const float4*)(X + (size_t)row_base * IN_DIM);
        #pragma unroll
        for (int it = 0; it < 8; ++it) {
            int i  = tid + it * 256;        // 0..2047
            int r  = i >> 7;                // 128 float4 per row
            int c4 = i & 127;
            float4 vl = xb[i];
            *(float4*)&xs[r * XS_STRIDE + c4 * 4] = vl;
        }
    }
    __syncthreads();

    // --- WMMA fragment coordinates (ISA 7.12.2 layouts) ---------------------
    const int m  = lane & 15;               // A: row within tile
    const int n  = lane & 15;               // B/C/D: column within tile
    const int kh = (lane >> 4) * 2;         // K half-select (0 or 2)

    const float* __restrict__ wcol = W + col_base + n;

    v8f acc = {};
    #pragma unroll 4
    for (int k = 0; k < KEY_DIM; k += 4) {
        // A fragment: lane holds X[m][k+kh .. k+kh+1]  (8B-aligned LDS read)
        v2f a = *(const v2f*)&xs[m * XS_STRIDE + k + kh];
        // B fragment: VGPR j holds W[k+kh+j][col_base+n]
        v2f b;
        b.x = wcol[(size_t)(k + kh + 0) * OUT_DIM];
        b.y = wcol[(size_t)(k + kh + 1) * OUT_DIM];
        acc = __builtin_amdgcn_wmma_f32_16x16x4_f32(
                  /*neg_a=*/false, a, /*neg_b=*/false, b,
                  /*c_mod=*/(short)0, acc,
                  /*reuse_a=*/false, /*reuse_b=*/false);
    }

    // --- epilogue: add bias, store D (VGPR g -> row g + 8*(lane>>4)) --------
    const float bv      = bias[col_base + n];
    const int   mrowoff = (lane >> 4) * 8;
    float* __restrict__ outb = Out + (size_t)row_base * OUT_DIM + col_base;
    #pragma unroll
    for (int g = 0; g < 8; ++g) {
        outb[(size_t)(g + mrowoff) * OUT_DIM + n] = acc[g] + bv;
    }
}

// ---------------------------------------------------------------------------
// Windowed attention: one block per (b, t_out).
// scores[f] = <q_center, k[4t+f]> * 1/sqrt(512); softmax over 32 taps;
// out[d] = sum_f attn[f] * v[4t+f][d].
// ---------------------------------------------------------------------------
__global__ __launch_bounds__(256) void window_attn_kernel(
    const float* __restrict__ Q,
    const float* __restrict__ K,
    const float* __restrict__ V,
    float* __restrict__ Out)
{
    const int bt  = blockIdx.x;
    const int b   = bt / T_OUT;
    const int t   = bt % T_OUT;
    const int tid = threadIdx.x;

    const int start  = t * STRIDE_T;
    const int center = start + FILT / 2;

    const float* __restrict__ qrow = Q + ((size_t)b * SEQ_T + center) * KEY_DIM;
    const float* __restrict__ kwin = K + ((size_t)b * SEQ_T + start)  * KEY_DIM;
    const float* __restrict__ vwin = V + ((size_t)b * SEQ_T + start)  * OUT_DIM;

    __shared__ float qsh[KEY_DIM];
    __shared__ float partial[256];
    __shared__ float attn[FILT];

    // stage q row
    #pragma unroll
    for (int i = tid; i < KEY_DIM; i += 256) qsh[i] = qrow[i];
    __syncthreads();

    // partial dot products: 8 threads per tap, 64 dims each
    const int f   = tid >> 3;   // 0..31
    const int seg = tid & 7;    // 0..7
    {
        const float* __restrict__ kr = kwin + (size_t)f * KEY_DIM + seg * 64;
        const float* __restrict__ qs = qsh + seg * 64;
        float p = 0.f;
        #pragma unroll 8
        for (int d = 0; d < 64; ++d) p += qs[d] * kr[d];
        partial[tid] = p;
    }
    __syncthreads();

    // reduce to 32 scores
    if (tid < FILT) {
        float s = 0.f;
        #pragma unroll
        for (int j = 0; j < 8; ++j) s += partial[tid * 8 + j];
        attn[tid] = s * 0.044194173824159216f;  // 1/sqrt(512)
    }
    __syncthreads();

    // softmax over 32 taps (serial in one lane; negligible cost)
    if (tid == 0) {
        float mx = attn[0];
        #pragma unroll
        for (int j = 1; j < FILT; ++j) mx = fmaxf(mx, attn[j]);
        float sum = 0.f;
        #pragma unroll
        for (int j = 0; j < FILT; ++j) { float e = expf(attn[j] - mx); attn[j] = e; sum += e; }
        float inv = 1.f / sum;
        #pragma unroll
        for (int j = 0; j < FILT; ++j) attn[j] *= inv;
    }
    __syncthreads();

    // weighted sum of v rows: 2 output dims per thread
    const int d0 = tid * 2;
    float acc0 = 0.f, acc1 = 0.f;
    #pragma unroll 4
    for (int fv = 0; fv < FILT; ++fv) {
        const float w = attn[fv];
        const float* __restrict__ vr = vwin + (size_t)fv * OUT_DIM + d0;
        acc0 += w * vr[0];
        acc1 += w * vr[1];
    }
    float* __restrict__ orow = Out + ((size_t)b * T_OUT + t) * OUT_DIM;
    orow[d0]     = acc0;
    orow[d0 + 1] = acc1;
}

// ---------------------------------------------------------------------------
// Launcher
// ---------------------------------------------------------------------------
extern "C" void kernel_launch(void* const* d_in, const int* in_sizes, int n_in,
                              void* d_out, int out_size, void* d_ws, size_t ws_size,
                              hipStream_t stream) {
    (void)in_sizes; (void)n_in; (void)out_size; (void)ws_size;

    const float* x  = (const float*)d_in[0];
    const float* Wq = (const float*)d_in[1];
    const float* bq = (const float*)d_in[2];
    const float* Wk = (const float*)d_in[3];
    const float* bk = (const float*)d_in[4];
    const float* Wv = (const float*)d_in[5];
    const float* bv = (const float*)d_in[6];
    float* out = (float*)d_out;

    const size_t plane = (size_t)MTOT * KEY_DIM;  // 16384*512 floats
    float* q = (float*)d_ws;
    float* k = q + plane;
    float* v = k + plane;

    dim3 gblk(256);
    dim3 ggrid(MTOT / 16, OUT_DIM / 128);   // 1024 x 4 blocks

    qkv_gemm_kernel<<<ggrid, gblk, 0, stream>>>(x, Wq, bq, q);
    qkv_gemm_kernel<<<ggrid, gblk, 0, stream>>>(x, Wk, bk, k);
    qkv_gemm_kernel<<<ggrid, gblk, 0, stream>>>(x, Wv, bv, v);

    window_attn_kernel<<<dim3(BATCH * T_OUT), dim3(256), 0, stream>>>(q, k, v, out);
}